// DSC_11536282157801
// MI455X (gfx1250) — compile-verified
//
#include <hip/hip_runtime.h>

typedef float v2f __attribute__((ext_vector_type(2)));
typedef float v8f __attribute__((ext_vector_type(8)));

#define B_TRIALS 512
#define L_HIST   128
#define P_OBS    256
#define M_CTRL   64
#define S_MAX    52          // lags s = 1..52
#define KW       625         // a*25 + l  (a,l in 0..24)
#define KWP      628         // padded to multiple of 4
#define ROWS_PAD 64          // 52 rows padded to 64
#define NW       16384       // 64*256 weight-GEMM N
#define KM       (S_MAX * P_OBS)   // 13312 main-GEMM K

// ---------------------------------------------------------------------------
// Kernel 0: build combined per-lag coefficient matrix Cfull[64][628]
//   row r -> lag s = r+1 ; col k = a*25 + l  (M's first two indices)
//   Cfull folds: Mk path (a==0), Ns path (G convolution of phis), Mj path.
//   Padding rows/cols are written as exact zeros.
// ---------------------------------------------------------------------------
__global__ void build_coef(const float* __restrict__ sigma,   // 25
                           const float* __restrict__ phi,     // 27 x 25
                           float* __restrict__ Cfull) {
    int idx = blockIdx.x * blockDim.x + threadIdx.x;          // 0..40191
    if (idx >= ROWS_PAD * KWP) return;
    int r = idx / KWP;
    int k = idx - r * KWP;
    float val = 0.0f;
    if (r < S_MAX && k < KW) {
        int s = r + 1;
        int a = k / 25;
        int l = k - a * 25;
        float Sl = sqrtf(sqrtf(sigma[l]));                    // sigma^0.25
        if (a == 0) {
            // Mk term: hidx = [1,1,2,...,26] -> lag 1 gets phi[0]+phi[1]
            if (s == 1)       val = Sl * (phi[0 * 25 + l] + phi[1 * 25 + l]);
            else if (s <= 26) val = Sl * phi[s * 25 + l];
        } else {
            int i = a - 1;                                    // 0..23 (< h)
            int q = s - 1;                                    // jj + kk = q
            int jj0 = q - 26; if (jj0 < 0) jj0 = 0;
            int jj1 = q;      if (jj1 > 25) jj1 = 25;
            float g = 0.0f;
            for (int jj = jj0; jj <= jj1; ++jj)
                g += phi[jj * 25 + i] * phi[(q - jj) * 25 + l];
            float Si = sqrtf(sqrtf(sigma[i]));
            val = Si * Sl * g;                                // Sprod * G
            if (s <= 25 && l == s - 1)                        // Mj diagonal term
                val += Si * phi[(s - 1) * 25 + i];
        }
    }
    Cfull[idx] = val;
}

// ---------------------------------------------------------------------------
// Kernel 1: weight GEMM  A[s, n] = sum_k Cfull[s,k] * Mflat[k, n]  (+ M_bar)
//   M = (64 rows padded), K = 625 (pad 628, B rows index-clamped), N = 16384.
//   One wave per 16x16 tile via V_WMMA_F32_16X16X4_F32.
//   Output scattered into the k-PAIRED layout consumed by the main GEMM:
//     AwTp[kappa>>1][c][kappa&1],  kappa = (52-s)*256 + p,  n = c*256+p
//   so each main-GEMM lane fetches its (k, k+1) B pair as one b64 load.
// ---------------------------------------------------------------------------
__global__ void weight_gemm(const float* __restrict__ Cfull,
                            const float* __restrict__ Mflat,  // 625 x 16384
                            const float* __restrict__ Mbar,   // 64 x 256
                            float* __restrict__ AwTp) {       // (13312/2) x 64 x 2
    const int lane = threadIdx.x & 31;
    const int wave = threadIdx.x >> 5;
    const int tile = blockIdx.x * 8 + wave;        // 0..4095
    const int rowTile = tile >> 10;                // 0..3
    const int colTile = tile & 1023;               // 0..1023
    const int half = lane >> 4;
    const int lm   = lane & 15;

    const int r = rowTile * 16 + lm;               // A row (0..63, all padded-valid)
    const int n = colTile * 16 + lm;               // B col

    v8f acc = {};
    for (int kt = 0; kt < KWP / 4; ++kt) {
        int ka = kt * 4 + 2 * half;
        v2f a = *(const v2f*)(Cfull + (size_t)r * KWP + ka);   // 8B-aligned
        int kb0 = ka;     if (kb0 > KW - 1) kb0 = KW - 1;      // clamp: coeff is 0 there
        int kb1 = ka + 1; if (kb1 > KW - 1) kb1 = KW - 1;
        v2f b;
        b.x = Mflat[(size_t)kb0 * NW + n];
        b.y = Mflat[(size_t)kb1 * NW + n];
        acc = __builtin_amdgcn_wmma_f32_16x16x4_f32(false, a, false, b,
                                                    (short)0, acc, false, false);
    }

    const int c = n >> 8;
    const int p = n & 255;
#pragma unroll
    for (int v = 0; v < 8; ++v) {
        int rr = rowTile * 16 + v + 8 * half;      // D row for this VGPR/lane-half
        if (rr < S_MAX) {
            int s = rr + 1;
            float val = acc[v];
            if (s == 1) val += Mbar[n];            // M_bar acts on lag 1 only
            int kappa = (S_MAX - s) * P_OBS + p;
            // paired layout: pair index, column, parity
            AwTp[(size_t)(kappa >> 1) * (2 * M_CTRL) + c * 2 + (kappa & 1)] = val;
        }
    }
}

// ---------------------------------------------------------------------------
// Kernel 2: main GEMM  u[b,c] = sum_kappa Ywin[b,kappa] * A[kappa,c]
//   M = 512, K = 13312, N = 64. Ywin is the contiguous last-52-step window:
//   Ywin[b,kappa] = Y[b*L*P + 76*P + kappa].
//   Block = 8 waves sharing one 16x16 output tile, K split 8 ways,
//   A and B operands are both single b64 loads per WMMA step,
//   LDS reduction, then full-tile store (covers all of d_out).
// ---------------------------------------------------------------------------
__global__ void main_gemm(const float* __restrict__ Y,
                          const float* __restrict__ AwTp,
                          float* __restrict__ out) {
    __shared__ float red[8 * 256];
    const int lane = threadIdx.x & 31;
    const int wave = threadIdx.x >> 5;             // K-chunk id 0..7
    const int bTile = blockIdx.x >> 2;             // 0..31
    const int cTile = blockIdx.x & 3;              // 0..3
    const int half = lane >> 4;
    const int lm   = lane & 15;

    const int brow = bTile * 16 + lm;
    const float* ybase = Y + (size_t)brow * (L_HIST * P_OBS) + 76 * P_OBS;
    const int ccol = cTile * 16 + lm;

    v8f acc = {};
    const int steps = (KM / 4) / 8;                // 416 exact
    const int kt0 = wave * steps;
    for (int kt = kt0; kt < kt0 + steps; ++kt) {
        int ka = kt * 4 + 2 * half;                // even
        v2f a = *(const v2f*)(ybase + ka);         // contiguous, 8B-aligned
        // paired layout: pair (ka>>1) holds k=ka (x) and k=ka+1 (y) for column ccol
        v2f b = *(const v2f*)(AwTp + (size_t)(ka >> 1) * (2 * M_CTRL) + ccol * 2);
        acc = __builtin_amdgcn_wmma_f32_16x16x4_f32(false, a, false, b,
                                                    (short)0, acc, false, false);
    }

#pragma unroll
    for (int v = 0; v < 8; ++v)
        red[wave * 256 + lane * 8 + v] = acc[v];
    __syncthreads();

    // 256 threads: each reduces one (lane, vgpr) slot across the 8 waves
    int t  = threadIdx.x;
    int rl = t >> 3;                               // original lane 0..31
    int rv = t & 7;                                // vgpr index 0..7
    float sum = 0.0f;
#pragma unroll
    for (int w = 0; w < 8; ++w) sum += red[w * 256 + rl * 8 + rv];
    int rhalf = rl >> 4, rlm = rl & 15;
    int b = bTile * 16 + rv + 8 * rhalf;
    int c = cTile * 16 + rlm;
    out[(size_t)b * M_CTRL + c] = sum;
}

// ---------------------------------------------------------------------------
extern "C" void kernel_launch(void* const* d_in, const int* in_sizes, int n_in,
                              void* d_out, int out_size, void* d_ws, size_t ws_size,
                              hipStream_t stream) {
    (void)in_sizes; (void)n_in; (void)out_size; (void)ws_size;
    const float* Y     = (const float*)d_in[0];   // 512 x 128 x 256
    const float* Mbar  = (const float*)d_in[1];   // 64 x 256
    const float* M     = (const float*)d_in[2];   // 25 x 25 x 64 x 256
    const float* sigma = (const float*)d_in[3];   // 25
    const float* phi   = (const float*)d_in[4];   // 27 x 25
    float* out = (float*)d_out;                   // 512 x 64

    float* Cfull = (float*)d_ws;                  // 64*628 floats
    float* AwTp  = Cfull + ROWS_PAD * KWP;        // 13312*64 floats (~3.4 MB total)

    build_coef<<<157, 256, 0, stream>>>(sigma, phi, Cfull);
    weight_gemm<<<512, 256, 0, stream>>>(Cfull, M, Mbar, AwTp);
    main_gemm<<<128, 256, 0, stream>>>(Y, AwTp, out);
}